// CropConLoss_89850715832378
// MI455X (gfx1250) — compile-verified
//
#include <hip/hip_runtime.h>
#include <math.h>

// ---------------------------------------------------------------------------
// CropConLoss on MI455X (gfx1250, wave32).
//
// Fused supervised-contrastive loss: never materialize the 8192x8192 sim
// matrix. Each wave computes 16x16 sim tiles with V_WMMA_F32_16X16X4_F32
// (native FP32 WMMA), applies exp/diag-mask/label-match in the epilogue and
// keeps only per-row running sums in VGPRs.
//
// v2 changes (reasoned from the v1 disassembly, no runtime available):
//  * dual WMMA accumulator chains (even/odd K-chunks) -> halves the
//    serial D->C dependency chain inside each 16x16 tile
//  * column sweep split 4-ways across blocks (512 blocks / ~2048 waves)
//    for SIMD occupancy; per-row partials combined via global_atomic_add_f32
// ---------------------------------------------------------------------------

typedef __attribute__((ext_vector_type(2))) float v2f;
typedef __attribute__((ext_vector_type(8))) float v8f;

#define M_TOT   8192
#define N_HALF  4096
#define D_DIM   128
#define C_CLS   64
#define TAU_INV 10.0f
#define LDSB_STRIDE 132   // 128 + 4 pad: lane n hits bank (4n+k)%64 -> conflict-free
#define NCHUNK  4         // column-sweep split factor
#define JT_PER_CHUNK (M_TOT / 16 / NCHUNK)   // 128 column tiles per chunk

// ---------------------------------------------------------------- utilities
// Zero the accumulation region: rowNum|rowDenR|rowDenP|clsCnt|lossAcc
// (contiguous in workspace, 3*8192 + 64 + 1 floats).
__global__ void cc_zero(float* accRegion, int n) {
    int i = blockIdx.x * blockDim.x + threadIdx.x;
    if (i < n) accRegion[i] = 0.0f;
}

// L2-normalize feats (concat proj2|proj3) and protos; emit concat labels.
// One wave per row: 4 floats/lane, wave32 butterfly for the squared norm.
__global__ __launch_bounds__(128) void cc_normalize(
    const float* __restrict__ proj2, const float* __restrict__ proj3,
    const int*   __restrict__ t2,    const int*   __restrict__ t3,
    const float* __restrict__ protosIn,
    float* __restrict__ feats, float* __restrict__ protosN,
    int* __restrict__ labels)
{
    const int lane = threadIdx.x & 31;
    const int wid  = threadIdx.x >> 5;
    const int row  = blockIdx.x * 4 + wid;             // 0 .. 8255
    const float* src;
    float* dst;
    if (row < N_HALF)            { src = proj2    +  row           * D_DIM; dst = feats   +  row           * D_DIM; }
    else if (row < M_TOT)        { src = proj3    + (row - N_HALF) * D_DIM; dst = feats   +  row           * D_DIM; }
    else if (row < M_TOT + C_CLS){ src = protosIn + (row - M_TOT)  * D_DIM; dst = protosN + (row - M_TOT)  * D_DIM; }
    else return;

    float4 x = *(const float4*)(src + lane * 4);
    float ss = x.x*x.x + x.y*x.y + x.z*x.z + x.w*x.w;
    #pragma unroll
    for (int m = 16; m >= 1; m >>= 1) ss += __shfl_xor(ss, m, 32);
    const float inv = 1.0f / fmaxf(sqrtf(ss), 1e-12f);
    x.x *= inv; x.y *= inv; x.z *= inv; x.w *= inv;
    *(float4*)(dst + lane * 4) = x;

    if (lane == 0 && row < M_TOT)
        labels[row] = (row < N_HALF) ? t2[row] : t3[row - N_HALF];
}

__global__ void cc_bincount(const int* __restrict__ labels, float* clsCnt) {
    int i = blockIdx.x * blockDim.x + threadIdx.x;
    if (i < M_TOT) atomicAdd(&clsCnt[labels[i]], 1.0f);
}

// ---------------------------------------------------------------- main fused
// Block = 128 threads = 4 waves; gridDim.x = (M/64) * NCHUNK.
// blockIdx.x = rowBlock * NCHUNK + chunk; wave w owns rows [rowBlock*64+w*16,+16)
// and sweeps column tiles [chunk*128, chunk*128+128). Chunk 0 also does protos.
// A fragment (16 rows x 128 K, FP32) pinned in 64 VGPRs/lane following the
// ISA 16x4 A layout: lanes 0-15 -> M=lane,K={0,1}; lanes 16-31 -> K={2,3}.
__global__ __launch_bounds__(128) void cc_main(
    const float* __restrict__ feats, const float* __restrict__ protosN,
    const int*   __restrict__ labels, const float* __restrict__ clsCnt,
    float* __restrict__ rowNum, float* __restrict__ rowDenR,
    float* __restrict__ rowDenP)
{
    __shared__ float ldsB[16 * LDSB_STRIDE];
    __shared__ int   ldsLab[16];
    __shared__ float ldsW[16];

    const int t        = threadIdx.x;
    const int lane     = t & 31;
    const int wid      = t >> 5;
    const int rowBlock = blockIdx.x / NCHUNK;
    const int chunk    = blockIdx.x % NCHUNK;
    const int rowBase  = rowBlock * 64 + wid * 16;
    const int nIdx     = lane & 15;   // A: row-in-tile, B/C/D: col-in-tile
    const int hi       = lane >> 4;   // half-wave selector (K-pair / row+8)

    // --- A fragments: this wave's 16 rows, K = 0..127 in 32 chunks of 4.
    v2f aF[32];
    {
        const float* rp = feats + (size_t)(rowBase + nIdx) * D_DIM;
        #pragma unroll
        for (int ch = 0; ch < 32; ++ch) {
            const int kk = ch * 4 + hi * 2;
            v2f a; a.x = rp[kk]; a.y = rp[kk + 1];
            aF[ch] = a;
        }
    }
    // Row labels for the 8 rows this lane's accumulator slots map to
    // (C/D layout: VGPR v, lanes 0-15 -> row v; lanes 16-31 -> row v+8).
    int rlab[8];
    #pragma unroll
    for (int v = 0; v < 8; ++v) rlab[v] = labels[rowBase + v + hi * 8];

    float accNum[8], accDenR[8], accDenP[8];
    #pragma unroll
    for (int v = 0; v < 8; ++v) { accNum[v] = 0.0f; accDenR[v] = 0.0f; accDenP[v] = 0.0f; }

    // ---------------- region pass: this chunk's 128 column tiles of feats
    const int jt0 = chunk * JT_PER_CHUNK;
    for (int jj = 0; jj < JT_PER_CHUNK; ++jj) {
        const int jt = jt0 + jj;
        __syncthreads();
        {
            const float* srcB = feats + (size_t)jt * 16 * D_DIM;
            for (int q = t; q < 512; q += 128) {                 // 16x128 in float4s
                const int r = q >> 5, k4 = (q & 31) << 2;
                *(float4*)(&ldsB[r * LDSB_STRIDE + k4]) =
                    *(const float4*)(srcB + r * D_DIM + k4);
            }
            if (t < 16) ldsLab[t] = labels[jt * 16 + t];
        }
        __syncthreads();

        // Dual accumulator chains: even chunks -> acc0, odd -> acc1.
        v8f acc0 = {}, acc1 = {};
        const float* bp = &ldsB[nIdx * LDSB_STRIDE + hi * 2];
        #pragma unroll
        for (int ch = 0; ch < 32; ch += 2) {
            v2f b0; b0.x = bp[ch * 4];     b0.y = bp[ch * 4 + 1];
            v2f b1; b1.x = bp[ch * 4 + 4]; b1.y = bp[ch * 4 + 5];
            acc0 = __builtin_amdgcn_wmma_f32_16x16x4_f32(
                false, aF[ch],     false, b0, (short)0, acc0, false, false);
            acc1 = __builtin_amdgcn_wmma_f32_16x16x4_f32(
                false, aF[ch + 1], false, b1, (short)0, acc1, false, false);
        }

        const int col  = jt * 16 + nIdx;
        const int clab = ldsLab[nIdx];
        #pragma unroll
        for (int v = 0; v < 8; ++v) {
            const int row = rowBase + v + hi * 8;
            float e = __expf((acc0[v] + acc1[v]) * TAU_INV);
            e = (row == col) ? 0.0f : e;                 // diagonal mask
            accDenR[v] += e;
            accNum[v]  += (rlab[v] == clab) ? e : 0.0f;  // same-label numerator
        }
    }

    // ---------------- proto pass (chunk 0 only): 4 tiles of prototypes
    if (chunk == 0) {
        for (int pt = 0; pt < C_CLS / 16; ++pt) {
            __syncthreads();
            {
                const float* srcB = protosN + (size_t)pt * 16 * D_DIM;
                for (int q = t; q < 512; q += 128) {
                    const int r = q >> 5, k4 = (q & 31) << 2;
                    *(float4*)(&ldsB[r * LDSB_STRIDE + k4]) =
                        *(const float4*)(srcB + r * D_DIM + k4);
                }
                if (t < 16) ldsW[t] = 1.0f / (clsCnt[pt * 16 + t] + 1.0f + 1e-6f);
            }
            __syncthreads();

            v8f acc0 = {}, acc1 = {};
            const float* bp = &ldsB[nIdx * LDSB_STRIDE + hi * 2];
            #pragma unroll
            for (int ch = 0; ch < 32; ch += 2) {
                v2f b0; b0.x = bp[ch * 4];     b0.y = bp[ch * 4 + 1];
                v2f b1; b1.x = bp[ch * 4 + 4]; b1.y = bp[ch * 4 + 5];
                acc0 = __builtin_amdgcn_wmma_f32_16x16x4_f32(
                    false, aF[ch],     false, b0, (short)0, acc0, false, false);
                acc1 = __builtin_amdgcn_wmma_f32_16x16x4_f32(
                    false, aF[ch + 1], false, b1, (short)0, acc1, false, false);
            }

            const int   c = pt * 16 + nIdx;
            const float w = ldsW[nIdx];
            #pragma unroll
            for (int v = 0; v < 8; ++v) {
                const float e = __expf((acc0[v] + acc1[v]) * TAU_INV);
                accDenP[v] += e * w;                      // 1/cls_freq weighted
                accNum[v]  += (rlab[v] == c) ? e : 0.0f;  // proto-at-label
            }
        }
    }

    // ---------------- 16-lane butterfly: per-row totals land in lanes 0 & 16
    #pragma unroll
    for (int v = 0; v < 8; ++v) {
        float n = accNum[v], dr = accDenR[v], dp = accDenP[v];
        #pragma unroll
        for (int m = 8; m >= 1; m >>= 1) {
            n  += __shfl_xor(n,  m, 32);
            dr += __shfl_xor(dr, m, 32);
            dp += __shfl_xor(dp, m, 32);
        }
        accNum[v] = n; accDenR[v] = dr; accDenP[v] = dp;
    }

    if (nIdx == 0) {
        #pragma unroll
        for (int v = 0; v < 8; ++v) {
            const int row = rowBase + v + hi * 8;
            atomicAdd(&rowNum[row],  accNum[v]);
            atomicAdd(&rowDenR[row], accDenR[v]);
            atomicAdd(&rowDenP[row], accDenP[v]);
        }
    }
}

// Per-row epilogue: loss_r = -log(num / (denR/freq + denP + eps)); wave-reduced
// partial sums accumulated into a single scalar.
__global__ __launch_bounds__(256) void cc_rowloss(
    const float* __restrict__ rowNum, const float* __restrict__ rowDenR,
    const float* __restrict__ rowDenP, const int* __restrict__ labels,
    const float* __restrict__ clsCnt, float* __restrict__ lossAcc)
{
    const int i = blockIdx.x * blockDim.x + threadIdx.x;
    float l = 0.0f;
    if (i < M_TOT) {
        const float fw    = clsCnt[labels[i]] + 1.0f + 1e-6f;
        const float denom = rowDenR[i] / fw + rowDenP[i] + 1e-12f;
        l = -__logf(rowNum[i] / denom);
    }
    #pragma unroll
    for (int m = 16; m >= 1; m >>= 1) l += __shfl_xor(l, m, 32);
    if ((threadIdx.x & 31) == 0) atomicAdd(lossAcc, l);
}

__global__ void cc_finalize(const float* lossAcc, float* out) {
    out[0] = lossAcc[0] * (1.0f / (float)M_TOT);
}

// ---------------------------------------------------------------------------
extern "C" void kernel_launch(void* const* d_in, const int* in_sizes, int n_in,
                              void* d_out, int out_size, void* d_ws, size_t ws_size,
                              hipStream_t stream) {
    const float* protos  = (const float*)d_in[0];   // [64,128]   f32
    const float* proj2   = (const float*)d_in[1];   // [4096,128] f32
    const int*   target2 = (const int*)  d_in[2];   // [4096]
    const float* proj3   = (const float*)d_in[3];   // [4096,128] f32
    const int*   target3 = (const int*)  d_in[4];   // [4096]
    float* out = (float*)d_out;

    // workspace layout (floats):
    //   feats_n | protos_n | labels(int) | [rowNum | rowDenR | rowDenP | clsCnt | lossAcc]
    // the bracketed tail is the zeroed accumulation region.
    float* ws      = (float*)d_ws;
    float* feats   = ws;                                        // 8192*128
    float* protosN = feats + (size_t)M_TOT * D_DIM;             // 64*128
    int*   labels  = (int*)(protosN + (size_t)C_CLS * D_DIM);   // 8192
    float* rowNum  = (float*)(labels + M_TOT);                  // 8192
    float* rowDenR = rowNum  + M_TOT;                           // 8192
    float* rowDenP = rowDenR + M_TOT;                           // 8192
    float* clsCnt  = rowDenP + M_TOT;                           // 64
    float* lossAcc = clsCnt + C_CLS;                            // 1

    const int nZero = 3 * M_TOT + C_CLS + 1;
    cc_zero     <<<(nZero + 255) / 256, 256, 0, stream>>>(rowNum, nZero);
    cc_normalize<<<(M_TOT + C_CLS) / 4, 128, 0, stream>>>(
        proj2, proj3, target2, target3, protos, feats, protosN, labels);
    cc_bincount <<<M_TOT / 256, 256, 0, stream>>>(labels, clsCnt);
    cc_main     <<<(M_TOT / 64) * NCHUNK, 128, 0, stream>>>(
        feats, protosN, labels, clsCnt, rowNum, rowDenR, rowDenP);
    cc_rowloss  <<<M_TOT / 256, 256, 0, stream>>>(
        rowNum, rowDenR, rowDenP, labels, clsCnt, lossAcc);
    cc_finalize <<<1, 1, 0, stream>>>(lossAcc, out);
}